// TripletLoss_19232863551583
// MI455X (gfx1250) — compile-verified
//
#include <hip/hip_runtime.h>

typedef float v2f __attribute__((ext_vector_type(2)));
typedef float v4f __attribute__((ext_vector_type(4)));
typedef float v8f __attribute__((ext_vector_type(8)));

#define DIMS   1024
#define TILE   16
#define MTILE  32            // rows per workgroup (2 WMMA strips)
#define JSPLIT 2             // workgroups sharing one row strip (split j range)
#define STR    2052          // combined interleaved row stride (floats), ≡4 mod 64
#define MARGIN 0.3f
#define EPS    1e-8f
#define FBIG   3.0e38f
#define UINF   0x7F7FFFFFu   // +FLT_MAX bit pattern

// ------------- kernel 1: row norms + global min-array init -------------
__global__ void __launch_bounds__(256)
tl_rowstats(const float* __restrict__ e, float* __restrict__ inv_norm,
            float* __restrict__ sqv, unsigned* __restrict__ posg,
            unsigned* __restrict__ negg, int n)
{
    const int row  = blockIdx.x * 8 + (threadIdx.x >> 5);
    const int lane = threadIdx.x & 31;
    if (row < n) {
        const v4f* p = (const v4f*)(e + (size_t)row * DIMS);
        float s = 0.0f;
        #pragma unroll
        for (int k = 0; k < DIMS / 4 / 32; ++k) {        // 8 x b128 loads per lane
            v4f x = p[lane + 32 * k];
            s += x.x * x.x + x.y * x.y + x.z * x.z + x.w * x.w;
        }
        #pragma unroll
        for (int off = 16; off >= 1; off >>= 1)
            s += __shfl_xor(s, off, 32);
        if (lane == 0) {
            float norm = sqrtf(s);
            float inv  = 1.0f / fmaxf(norm, 1e-12f);
            inv_norm[row] = inv;
            float t = norm * inv;                        // == 1 unless norm < 1e-12
            sqv[row] = t * t;                            // diag of normalized Gram
        }
    }
    const int gi = blockIdx.x * 256 + threadIdx.x;       // reset per-row global mins
    if (gi < n) { posg[gi] = UINF; negg[gi] = UINF; }
}

// ---- kernel 2: fused f32-WMMA Gram tiles -> masked d^2 row-min (register-resident) ----
__global__ void __launch_bounds__(256)
tl_hardmine(const float* __restrict__ e, const long long* __restrict__ labels,
            const float* __restrict__ inv_norm, const float* __restrict__ sqv,
            unsigned* __restrict__ posg, unsigned* __restrict__ negg, int n)
{
    // Two normalized 16x1024 strips, k-interleaved in groups of 4 floats:
    // row m = [s0 k0..3][s1 k0..3][s0 k4..7][s1 k4..7]...  (~128 KB)
    __shared__ float Als[TILE * STR];
    __shared__ int   labI[MTILE];
    __shared__ float sqI[MTILE];

    const int tid   = threadIdx.x;
    const int lane  = tid & 31;
    const int wave  = tid >> 5;
    const int strip = blockIdx.x / JSPLIT;      // which 32-row band
    const int jhalf = blockIdx.x % JSPLIT;      // which half of the j range
    const int ibase = strip * MTILE;

    // Stage: idx -> (m, s, g); consecutive tid walk g => coalesced global reads.
    for (int idx = tid; idx < TILE * 2 * (DIMS / 4); idx += 256) {
        const int g = idx & 255;                // k-group of 4 floats
        const int s = (idx >> 8) & 1;           // strip
        const int m = idx >> 9;                 // row within strip
        const int row = ibase + s * TILE + m;
        v4f x = *(const v4f*)(e + (size_t)row * DIMS + g * 4);
        x *= inv_norm[row];
        *(v4f*)(&Als[m * STR + g * 8 + s * 4]) = x;
    }
    if (tid < MTILE) {
        labI[tid] = ((const int*)labels)[2 * (ibase + tid)];  // values < 64: low dword
        sqI[tid]  = sqv[ibase + tid];
    }
    __syncthreads();

    // WMMA f32 16x16x4 fragment mapping (wave32):
    //   A: lane%16 = row M, lanes>=16 take K+2..3; VGPR pair = K, K+1
    //   B: lane%16 = col N, same K split; C: vgpr r -> row r (+8 for lanes>=16)
    // Lane l's C columns are n = l&15 == the row j that lane loaded for B, so
    // B normalization commutes out of the K loop: dot_hat = (e_hat_i . e_j) * inv_j.
    const int    sub   = lane & 15;
    const int    khalf = (lane >> 4) << 1;      // 0 or 2
    const int    mhi   = (lane >> 4) << 3;      // 0 or 8
    const float* aBase = &Als[sub * STR + khalf];  // strip0 frag; strip1 at +4 floats

    // Hoisted per-lane row constants (8 C rows per strip).
    float sqA[8], sqB[8]; int labA[8], labB[8];
    #pragma unroll
    for (int r = 0; r < 8; ++r) {
        sqA[r]  = sqI[mhi + r];         labA[r] = labI[mhi + r];
        sqB[r]  = sqI[TILE + mhi + r];  labB[r] = labI[TILE + mhi + r];
    }

    // Register-resident masked d^2 running minima (sqrt/relu are monotone).
    float pd0[8], nd0[8], pd1[8], nd1[8];
    #pragma unroll
    for (int r = 0; r < 8; ++r) { pd0[r] = nd0[r] = pd1[r] = nd1[r] = FBIG; }

    const int ntiles = n / TILE;
    const int jt0    = jhalf * (ntiles / JSPLIT);
    const int jt1    = jt0 + ntiles / JSPLIT;
    for (int jt = jt0 + wave; jt < jt1; jt += 8) {
        const int    j      = jt * TILE + sub;
        const float  bscale = inv_norm[j];
        const float  sqj    = sqv[j];
        const int    labj   = ((const int*)labels)[2 * j];
        const float* bptr   = e + (size_t)j * DIMS + khalf;
        __builtin_prefetch(bptr, 0, 3);

        v8f c0 = {0.0f, 0.0f, 0.0f, 0.0f, 0.0f, 0.0f, 0.0f, 0.0f};
        v8f c1 = {0.0f, 0.0f, 0.0f, 0.0f, 0.0f, 0.0f, 0.0f, 0.0f};
        #pragma unroll 8
        for (int g = 0; g < DIMS / 4; ++g) {     // one 4-wide K step per g
            v2f b  = *(const v2f*)(bptr + g * 4);        // global b64, feeds 2 WMMAs
            v2f a0 = *(const v2f*)(aBase + g * 8);       // fused ds_load_2addr_b64
            v2f a1 = *(const v2f*)(aBase + g * 8 + 4);   //   (16 B apart)
            c0 = __builtin_amdgcn_wmma_f32_16x16x4_f32(
                     false, a0, false, b, (short)0, c0, false, false);
            c1 = __builtin_amdgcn_wmma_f32_16x16x4_f32(
                     false, a1, false, b, (short)0, c1, false, false);
        }

        #pragma unroll
        for (int r = 0; r < 8; ++r) {
            const float d20 = sqA[r] + sqj - 2.0f * (bscale * c0[r]);
            const float d21 = sqB[r] + sqj - 2.0f * (bscale * c1[r]);
            const bool s0 = (labA[r] == labj);
            const bool s1 = (labB[r] == labj);
            pd0[r] = fminf(pd0[r], s0 ? d20 : FBIG);
            nd0[r] = fminf(nd0[r], s0 ? FBIG : d20);
            pd1[r] = fminf(pd1[r], s1 ? d21 : FBIG);
            nd1[r] = fminf(nd1[r], s1 ? FBIG : d21);
        }
    }

    // One cross-lane reduction per wave (not per tile), then global atomicMin
    // on the monotone u32 encoding of relu(d^2)+eps.
    #pragma unroll
    for (int r = 0; r < 8; ++r) {
        float p0 = pd0[r], n0 = nd0[r], p1 = pd1[r], n1 = nd1[r];
        #pragma unroll
        for (int off = 8; off >= 1; off >>= 1) {
            p0 = fminf(p0, __shfl_xor(p0, off, 16));
            n0 = fminf(n0, __shfl_xor(n0, off, 16));
            p1 = fminf(p1, __shfl_xor(p1, off, 16));
            n1 = fminf(n1, __shfl_xor(n1, off, 16));
        }
        if (sub == 0) {
            const int i0 = ibase + mhi + r;
            const int i1 = i0 + TILE;
            atomicMin(&posg[i0], __float_as_uint(fmaxf(p0, 0.0f) + EPS));
            atomicMin(&negg[i0], __float_as_uint(fmaxf(n0, 0.0f) + EPS));
            atomicMin(&posg[i1], __float_as_uint(fmaxf(p1, 0.0f) + EPS));
            atomicMin(&negg[i1], __float_as_uint(fmaxf(n1, 0.0f) + EPS));
        }
    }
}

// -------------------- kernel 3: sqrt + loss + mean (single block) --------------------
__global__ void __launch_bounds__(256)
tl_finalize(const unsigned* __restrict__ posg, const unsigned* __restrict__ negg,
            float* __restrict__ out, int n)
{
    __shared__ float stot[8], scnt[8];
    float total = 0.0f, cnt = 0.0f;
    for (int i = threadIdx.x; i < n; i += 256) {
        const float hp = sqrtf(__uint_as_float(posg[i]));   // relu+eps pre-applied
        const float hn = sqrtf(__uint_as_float(negg[i]));
        const float loss = fmaxf(hp - hn + MARGIN, 0.0f);
        if (loss > 0.0f) { total += loss; cnt += 1.0f; }
    }
    #pragma unroll
    for (int off = 16; off >= 1; off >>= 1) {
        total += __shfl_xor(total, off, 32);
        cnt   += __shfl_xor(cnt,   off, 32);
    }
    const int wave = threadIdx.x >> 5;
    if ((threadIdx.x & 31) == 0) { stot[wave] = total; scnt[wave] = cnt; }
    __syncthreads();
    if (threadIdx.x == 0) {
        float t = 0.0f, c = 0.0f;
        #pragma unroll
        for (int w = 0; w < 8; ++w) { t += stot[w]; c += scnt[w]; }
        out[0] = (c > 0.0f) ? (t / fmaxf(c, 1.0f)) : 0.0f;
    }
}

extern "C" void kernel_launch(void* const* d_in, const int* in_sizes, int n_in,
                              void* d_out, int out_size, void* d_ws, size_t ws_size,
                              hipStream_t stream)
{
    const float*     emb    = (const float*)d_in[0];     // [n, 1024] f32
    const long long* labels = (const long long*)d_in[1]; // [n] i64
    const int n = in_sizes[1];                           // 4096

    float*    inv_norm = (float*)d_ws;        // n floats
    float*    sqv      = inv_norm + n;        // n floats
    unsigned* posg     = (unsigned*)(sqv + n);// n u32 (encoded relu(d2)+eps)
    unsigned* negg     = posg + n;            // n u32

    tl_rowstats<<<(n + 7) / 8, 256, 0, stream>>>(emb, inv_norm, sqv, posg, negg, n);
    tl_hardmine<<<(n / MTILE) * JSPLIT, 256, 0, stream>>>(emb, labels, inv_norm, sqv,
                                                          posg, negg, n);
    tl_finalize<<<1, 256, 0, stream>>>(posg, negg, (float*)d_out, n);
}